// BatchBlur_SV2_4105988735744
// MI455X (gfx1250) — compile-verified
//
#include <hip/hip_runtime.h>

// ---- problem constants (from reference) ----
#define LK    19
#define PADR  9
#define K2    361
#define Hdim  256
#define Wdim  256
#define Bdim  2
#define Cdim  3

// ---- tiling ----
#define TW       32            // output tile width (pixels)
#define TH       8             // output tile height
#define NPX      2             // pixels per thread along j
#define TXN      (TW / NPX)    // 16 threads along j
#define NTHREADS (TXN * TH)    // 128 threads = 4 waves
#define XW       (TW + LK - 1) // 50: padded tile width
#define XH       (TH + LK - 1) // 26: padded tile height
#define XSTRIDE  51            // odd stride -> adjacent LDS rows hit disjoint bank parity

typedef __attribute__((ext_vector_type(2))) float v2f;
typedef __attribute__((address_space(1))) int as1_int;  // global
typedef __attribute__((address_space(3))) int as3_int;  // LDS

// gfx1250 async global->LDS builtin (confirmed present by round-1 diagnostic:
// signature takes non-const AS(1) int* src, AS(3) int* dst, imm offset, imm cpol)
#if defined(__has_builtin)
#  if __has_builtin(__builtin_amdgcn_global_load_async_to_lds_b32)
#    define HAVE_ASYNC_LDS 1
#  endif
#endif

__device__ __forceinline__ int reflect_idx(int t) {
  // jnp.pad 'reflect' (mirror excluding edge); |t| <= 264 so one reflection suffices
  t = (t < 0) ? -t : t;
  t = (t > Hdim - 1) ? (2 * (Hdim - 1) - t) : t;
  return t;
}

__global__ __launch_bounds__(NTHREADS) void svblur_kernel(
    const float* __restrict__ x,    // [B, C, H, W]
    const float* __restrict__ ker,  // [B, K2, H, W]
    float* __restrict__ out)        // [B, C, H, W]
{
  __shared__ float xs[Cdim * XH * XSTRIDE];

  const int b   = blockIdx.z;
  const int j0  = blockIdx.x * TW;
  const int i0  = blockIdx.y * TH;
  const int tid = threadIdx.x;

  // ---- stage reflect-padded x tile (3 channels) into LDS ----
  {
    const int total = Cdim * XH * XW;
    for (int t = tid; t < total; t += NTHREADS) {
      const int c  = t / (XH * XW);
      const int r  = t - c * (XH * XW);
      const int py = r / XW;
      const int px = r - py * XW;
      const int gy = reflect_idx(i0 + py - PADR);
      const int gx = reflect_idx(j0 + px - PADR);
      const float* src = x + ((((size_t)b * Cdim + c) * Hdim + gy) * Wdim + gx);
      float* dst = &xs[(c * XH + py) * XSTRIDE + px];
#if defined(HAVE_ASYNC_LDS)
      __builtin_amdgcn_global_load_async_to_lds_b32(
          (as1_int*)src, (as3_int*)dst, 0, 0);
#else
      *dst = *src;
#endif
    }
#if defined(HAVE_ASYNC_LDS)
#  if __has_builtin(__builtin_amdgcn_s_wait_asynccnt)
    __builtin_amdgcn_s_wait_asynccnt(0);
#  else
    asm volatile("s_wait_asynccnt 0" ::: "memory");
#  endif
#endif
    __syncthreads();
  }

  // ---- per-thread: 2 adjacent output pixels, 3 channels ----
  const int tx = tid % TXN;      // 0..15
  const int ty = tid / TXN;      // 0..7
  const int lx = tx * NPX;       // even local x of first pixel
  const int gi = i0 + ty;
  const int gj = j0 + lx;

  const float* kp =
      ker + (((size_t)b * K2) * Hdim + gi) * (size_t)Wdim + gj;

  float acc00 = 0.f, acc01 = 0.f;   // channel 0, pixels {gj, gj+1}
  float acc10 = 0.f, acc11 = 0.f;   // channel 1
  float acc20 = 0.f, acc21 = 0.f;   // channel 2

  for (int u = 0; u < LK; ++u) {
    const float* r0 = &xs[(0 * XH + ty + u) * XSTRIDE + lx];
    const float* r1 = &xs[(1 * XH + ty + u) * XSTRIDE + lx];
    const float* r2 = &xs[(2 * XH + ty + u) * XSTRIDE + lx];
    const float* kpu = kp + (size_t)u * LK * (Hdim * Wdim);
#pragma unroll
    for (int v = 0; v < LK; ++v) {
      // one-pass 189MB stream: non-temporal so it doesn't evict x / caches
      const v2f kv = __builtin_nontemporal_load(
          (const v2f*)(kpu + (size_t)v * (Hdim * Wdim)));
      const float a0 = r0[v], b0 = r0[v + 1];  // b0 CSEs into next iter's a0
      const float a1 = r1[v], b1 = r1[v + 1];
      const float a2 = r2[v], b2 = r2[v + 1];
      acc00 += a0 * kv.x;  acc01 += b0 * kv.y;
      acc10 += a1 * kv.x;  acc11 += b1 * kv.y;
      acc20 += a2 * kv.x;  acc21 += b2 * kv.y;
    }
  }

  const float inv = 1.0f / (float)K2;
  float* o0 = out + ((((size_t)b * Cdim + 0) * Hdim + gi) * Wdim + gj);
  float* o1 = out + ((((size_t)b * Cdim + 1) * Hdim + gi) * Wdim + gj);
  float* o2 = out + ((((size_t)b * Cdim + 2) * Hdim + gi) * Wdim + gj);
  o0[0] = acc00 * inv;  o0[1] = acc01 * inv;
  o1[0] = acc10 * inv;  o1[1] = acc11 * inv;
  o2[0] = acc20 * inv;  o2[1] = acc21 * inv;
}

extern "C" void kernel_launch(void* const* d_in, const int* in_sizes, int n_in,
                              void* d_out, int out_size, void* d_ws, size_t ws_size,
                              hipStream_t stream) {
  const float* x   = (const float*)d_in[0];   // [2,3,256,256] f32
  const float* ker = (const float*)d_in[1];   // [2,361,256,256] f32
  float* out       = (float*)d_out;           // [2,3,256,256] f32
  (void)in_sizes; (void)n_in; (void)out_size; (void)d_ws; (void)ws_size;

  dim3 grid(Wdim / TW, Hdim / TH, Bdim);      // 8 x 32 x 2 = 512 blocks
  svblur_kernel<<<grid, NTHREADS, 0, stream>>>(x, ker, out);
}